// PCgraph_28827820490922
// MI455X (gfx1250) — compile-verified
//
#include <hip/hip_runtime.h>
#include <math.h>
#include <stdint.h>

// ---------------------------------------------------------------------------
// Predictive-coding graph relaxation, reformulated as dense f32 WMMA GEMMs.
//   mu  = tanh(x) @ M1^T      (M1[s,t] = sum of w_e over edges s<-t)
//   agg = err     @ M1
// M=128 (batch), N=K=2048 (2000 padded). M1 built once per launch from the
// edge list with float atomics; 16MB -> resident in the 192MB L2 across all
// 20 GEMMs. WMMA: V_WMMA_F32_16X16X4_F32 (full fp32 preserves the fixed
// point). B tiles staged via GLOBAL_LOAD_ASYNC_TO_LDS_B128 (ASYNCcnt) into a
// double-buffered LDS tile; A/B fragments software-pipelined one k-step deep
// so loads overlap the WMMA stream instead of hard-waiting.
// Workspace: ~37 MB (2x 16MB matrices + 3x 1MB activation buffers).
// ---------------------------------------------------------------------------

#define NP      2048      // padded node count (N=2000)
#define LRX     0.5f
#define T_ITERS 10
#define BN      64        // N-columns per workgroup
#define BK      32        // K-depth per LDS stage
#define LDSB    72        // padded LDS row stride (floats)

typedef __attribute__((ext_vector_type(2))) float v2f;
typedef __attribute__((ext_vector_type(8))) float v8f;
typedef __attribute__((__vector_size__(4 * sizeof(int)))) int v4i_t;

#if defined(__has_builtin)
#if __has_builtin(__builtin_amdgcn_global_load_async_to_lds_b128) && \
    __has_builtin(__builtin_amdgcn_s_wait_asynccnt)
#define USE_ASYNC_LDS 1
#endif
#endif

#ifdef USE_ASYNC_LDS
// Builtin signature: (v4i addrspace(1)* gsrc, v4i addrspace(3)* lds, Ii, Ii).
// flat->LDS: low 32 bits of a flat LDS address are the LDS byte offset.
#define AS1P(p) ((__attribute__((address_space(1))) v4i_t*)(uintptr_t)(p))
#define AS3P(p) ((__attribute__((address_space(3))) v4i_t*)(uint32_t)(uintptr_t)(p))
#endif

__global__ void zero_kernel(float* __restrict__ p, long long n) {
  long long i = (long long)blockIdx.x * blockDim.x + threadIdx.x;
  long long stride = (long long)gridDim.x * blockDim.x;
  for (; i < n; i += stride) p[i] = 0.0f;
}

// Scatter edge weights into dense M1 [s][t] and its transpose M1T [t][s].
__global__ void build_mats(const int* __restrict__ ei, const float* __restrict__ w,
                           float* __restrict__ M1, float* __restrict__ M1T, int E) {
  int e = blockIdx.x * blockDim.x + threadIdx.x;
  if (e >= E) return;
  int s = ei[e];        // row 0: source
  int t = ei[E + e];    // row 1: target
  float wv = w[e];
  atomicAdd(&M1[(long long)s * NP + t], wv);
  atomicAdd(&M1T[(long long)t * NP + s], wv);
}

// xb [B,NP] = padded copy of x ; fx = tanh(xb)
__global__ void init_x(const float* __restrict__ xin, float* __restrict__ xb,
                       float* __restrict__ fx, int B, int N) {
  int i = blockIdx.x * blockDim.x + threadIdx.x;
  if (i >= B * NP) return;
  int b = i / NP, n = i % NP;
  float v = (n < N) ? xin[b * N + n] : 0.0f;
  xb[i] = v;
  fx[i] = tanhf(v);
}

__global__ void copy_out(const float* __restrict__ xb, float* __restrict__ out,
                         int B, int N) {
  int i = blockIdx.x * blockDim.x + threadIdx.x;
  if (i >= B * N) return;
  int b = i / N, n = i % N;
  out[i] = xb[(long long)b * NP + n];
}

// Fused GEMM: C = A(128xNP) * Bm(NPxNP), epilogue depends on mode.
//  mode 1: out0(err) = (xb - C) * mask            (C = mu)
//  mode 2: out0(xb)  = xb - LRX*(A - (1-fx^2)*C)*mask ; out1(fx) = tanh(out0)
// (in mode 2, A is the err buffer; out0 aliases xb, out1 aliases fx —
//  each element is read+written by its own thread only.)
__global__ __launch_bounds__(256) void gemm_fused(
    const float* __restrict__ A, const float* __restrict__ Bm,
    const float* xb, const float* fxin, const float* __restrict__ mask,
    float* out0, float* out1, int N, int mode) {
  __shared__ float Bs[2][BK * LDSB];   // double-buffered B tile

  const int tid  = threadIdx.x;
  const int wave = tid >> 5;          // 8 waves, wave32
  const int lane = tid & 31;
  const int ml   = lane & 15;         // matrix-minor lane index
  const int h    = lane >> 4;         // half-wave selector
  const int m0   = wave * 16;         // this wave's M rows [m0, m0+16)
  const int n0   = blockIdx.x * BN;   // this WG's N columns

  const int srow = tid >> 4;          // staging: row 0..15 (+16 on 2nd rep)
  const int sc4  = (tid & 15) << 2;   // staging: column (float4 granules)

  v8f acc[4];
  acc[0] = {}; acc[1] = {}; acc[2] = {}; acc[3] = {};

  const float* Arow = A + (long long)(m0 + ml) * NP;  // per-lane A row

  // stage B tile [BK x BN] for chunk k0 into buffer `buf`
  auto stage = [&](int k0, int buf) {
#pragma unroll
    for (int rep = 0; rep < 2; ++rep) {
      int row = srow + rep * 16;
      const float* g = &Bm[(long long)(k0 + row) * NP + n0 + sc4];
      float* l = &Bs[buf][row * LDSB + sc4];
#ifdef USE_ASYNC_LDS
      __builtin_amdgcn_global_load_async_to_lds_b128(AS1P(g), AS3P(l), 0, 0);
#else
      *(float4*)l = *(const float4*)g;
#endif
    }
  };

  stage(0, 0);
#ifdef USE_ASYNC_LDS
  __builtin_amdgcn_s_wait_asynccnt(0);
#endif
  __syncthreads();

  for (int k0 = 0; k0 < NP; k0 += BK) {
    const int buf = (k0 >> 5) & 1;
    // overlap: stage next chunk into the other buffer while computing
    if (k0 + BK < NP) stage(k0 + BK, buf ^ 1);
    // pull chunk k0+2*BK toward L2 (global_prefetch_b8)
    if (k0 + 2 * BK < NP && tid < BK)
      __builtin_prefetch(&Bm[(long long)(k0 + 2 * BK + tid) * NP + n0], 0, 1);

    const float* Bsb = &Bs[buf][0];

    // ---- 2-deep software pipeline over the 8 k-steps of this chunk ----
    // A fragment (16x4 f32): v0=K(2h), v1=K(2h+1), row = m0+ml
    // B fragment (4x16 f32): v0=B[kk+2h][n], v1=B[kk+2h+1][n]
    v2f a_cur = *(const v2f*)(&Arow[k0 + 2 * h]);
    v2f b_cur[4];
#pragma unroll
    for (int nt = 0; nt < 4; ++nt) {
      b_cur[nt][0] = Bsb[(2 * h)     * LDSB + nt * 16 + ml];
      b_cur[nt][1] = Bsb[(2 * h + 1) * LDSB + nt * 16 + ml];
    }

#pragma unroll
    for (int q = 0; q < 8; ++q) {
      v2f a_nxt = a_cur;
      v2f b_nxt[4];
      if (q < 7) {
        const int kk = (q + 1) * 4;
        a_nxt = *(const v2f*)(&Arow[k0 + kk + 2 * h]);
#pragma unroll
        for (int nt = 0; nt < 4; ++nt) {
          b_nxt[nt][0] = Bsb[(kk + 2 * h)     * LDSB + nt * 16 + ml];
          b_nxt[nt][1] = Bsb[(kk + 2 * h + 1) * LDSB + nt * 16 + ml];
        }
      } else {
#pragma unroll
        for (int nt = 0; nt < 4; ++nt) b_nxt[nt] = b_cur[nt];
      }

#pragma unroll
      for (int nt = 0; nt < 4; ++nt)
        acc[nt] = __builtin_amdgcn_wmma_f32_16x16x4_f32(
            /*neg_a=*/false, a_cur, /*neg_b=*/false, b_cur[nt],
            /*c_mod=*/(short)0, acc[nt], /*reuse_a=*/false, /*reuse_b=*/false);

      a_cur = a_nxt;
#pragma unroll
      for (int nt = 0; nt < 4; ++nt) b_cur[nt] = b_nxt[nt];
    }

#ifdef USE_ASYNC_LDS
    __builtin_amdgcn_s_wait_asynccnt(0);   // staged tile fully in LDS
#endif
    __syncthreads();
  }

  // ---- epilogue: C/D layout -> element (m0 + r + 8h, n0 + nt*16 + ml) ----
#pragma unroll
  for (int nt = 0; nt < 4; ++nt) {
    int gn = n0 + nt * 16 + ml;
    float mval = (gn < N) ? mask[gn] : 0.0f;
#pragma unroll
    for (int r = 0; r < 8; ++r) {
      int m = m0 + r + 8 * h;
      long long idx = (long long)m * NP + gn;
      float cv = acc[nt][r];
      if (mode == 1) {
        out0[idx] = (xb[idx] - cv) * mval;               // err
      } else {
        float xc   = xb[idx];
        float fv   = fxin[idx];
        float dx   = A[idx] - (1.0f - fv * fv) * cv;     // err - dfdx*agg
        float xn   = xc - LRX * dx * mval;
        out0[idx]  = xn;                                  // xb (in place)
        out1[idx]  = tanhf(xn);                           // fx for next step
      }
    }
  }
}

extern "C" void kernel_launch(void* const* d_in, const int* in_sizes, int n_in,
                              void* d_out, int out_size, void* d_ws, size_t ws_size,
                              hipStream_t stream) {
  const float* x_in = (const float*)d_in[0];   // [B, N]
  const int*   ei   = (const int*)d_in[1];     // [2, E]
  const float* w    = (const float*)d_in[2];   // [E]
  const float* mask = (const float*)d_in[3];   // [N]

  const int N = in_sizes[3];          // 2000
  const int B = in_sizes[0] / N;      // 128
  const int E = in_sizes[1] / 2;      // 131072

  // workspace layout (floats): M1 | M1T | xb | fx | err   (~37 MB total)
  float* M1  = (float*)d_ws;
  float* M1T = M1  + (size_t)NP * NP;
  float* xb  = M1T + (size_t)NP * NP;
  float* fx  = xb  + (size_t)B * NP;
  float* err = fx  + (size_t)B * NP;

  zero_kernel<<<2048, 256, 0, stream>>>(M1, 2LL * NP * NP);
  build_mats<<<(E + 255) / 256, 256, 0, stream>>>(ei, w, M1, M1T, E);
  init_x<<<(B * NP + 255) / 256, 256, 0, stream>>>(x_in, xb, fx, B, N);

  dim3 grid(NP / BN);   // 32 workgroups per GEMM
  for (int t = 0; t < T_ITERS; ++t) {
    // mu = fx @ M1T ; err = (xb - mu)*mask
    gemm_fused<<<grid, 256, 0, stream>>>(fx,  M1T, xb, fx, mask, err, nullptr, N, 1);
    // agg = err @ M1 ; xb -= LRX*(err - dfdx*agg)*mask ; fx = tanh(xb)
    gemm_fused<<<grid, 256, 0, stream>>>(err, M1,  xb, fx, mask, xb,  fx,      N, 2);
  }

  copy_out<<<(B * N + 255) / 256, 256, 0, stream>>>(xb, (float*)d_out, B, N);
}